// MMTGNN_74818330296383
// MI455X (gfx1250) — compile-verified
//
#include <hip/hip_runtime.h>
#include <hip/hip_bf16.h>
#include <math.h>

// ---------------- problem constants (from reference) ----------------
#define B_   128
#define T_   8
#define P_   49
#define H_   64
#define NH_  4
#define HD_  16
#define N_   343               // (T-1)*P
#define NP_  352               // padded to 22*16 so WMMA tiles need no guards
#define NT_  22                // NP_/16
#define BN_  (B_*N_)           // real rows
#define NC_  2
#define THRESH_    0.6f
#define NEG_SLOPE_ 0.2f

typedef __attribute__((ext_vector_type(2))) float v2f;
typedef __attribute__((ext_vector_type(8))) float v8f;

__device__ __forceinline__ float lrelu(float x) { return x > 0.f ? x : NEG_SLOPE_ * x; }

// fp32 WMMA: D = A(16x4) * B(4x16) + C, one wave per tile.
__device__ __forceinline__ v8f wmma_f32(v2f a, v2f b, v8f c) {
    return __builtin_amdgcn_wmma_f32_16x16x4_f32(false, a, false, b, (short)0, c, false, false);
}

// ---------------- init: zero pad rows + stats arrays (once per call) ----------------
__global__ __launch_bounds__(256) void k_initpad(float* __restrict__ node,
                                                 float* __restrict__ xfn,
                                                 float* __restrict__ xtn,
                                                 float* __restrict__ out1,
                                                 float* __restrict__ ssrc,
                                                 float* __restrict__ sdst,
                                                 float* __restrict__ mrow,
                                                 float* __restrict__ rden) {
    int idx = blockIdx.x * 256 + threadIdx.x;
    const int padElems = B_ * (NP_ - N_) * 64;     // 73728
    if (idx < padElems) {
        int b = idx / ((NP_ - N_) * 64);
        int r = idx % ((NP_ - N_) * 64);
        size_t off = ((size_t)b * NP_ + N_) * 64 + r;
        node[off] = 0.f; xfn[off] = 0.f; xtn[off] = 0.f; out1[off] = 0.f;
    }
    if (idx < B_ * NP_) {
        ssrc[idx] = 0.f; sdst[idx] = 0.f; mrow[idx] = 0.f; rden[idx] = 0.f;
    }
}

// ---------------- kernel 1: edge MLP -> node features ----------------
__global__ __launch_bounds__(64) void k_node(const float* __restrict__ reps,
                                             const float* __restrict__ Wm1,
                                             const float* __restrict__ bm1,
                                             const float* __restrict__ Wm2,
                                             const float* __restrict__ bm2,
                                             float* __restrict__ node) {
    __shared__ float cat[128];
    __shared__ float h1[64];
    int blk = blockIdx.x;               // b*N_ + n (real rows only)
    int b = blk / N_, n = blk % N_;
    int t = n / P_, p = n % P_;
    int tid = threadIdx.x;
    const float* rf = reps + ((size_t)((b * T_ + t) * P_ + p)) * H_;
    const float* rt = reps + ((size_t)((b * T_ + t + 1) * P_ + p)) * H_;
    cat[tid]      = rf[tid];
    cat[64 + tid] = rt[tid];
    __syncthreads();
    float acc = bm1[tid];
    const float* w = Wm1 + tid * 128;
#pragma unroll 8
    for (int k = 0; k < 128; ++k) acc += cat[k] * w[k];
    h1[tid] = fmaxf(acc, 0.f);
    __syncthreads();
    float acc2 = bm2[tid];
    const float* w2 = Wm2 + tid * 64;
#pragma unroll 8
    for (int k = 0; k < 64; ++k) acc2 += h1[k] * w2[k];
    node[((size_t)b * NP_ + n) * 64 + tid] = acc2;
}

// ---------------- kernel 2: l2-normalize X_f / X_t rows ----------------
__global__ __launch_bounds__(64) void k_norm(const float* __restrict__ reps,
                                             float* __restrict__ xfn,
                                             float* __restrict__ xtn) {
    __shared__ float red[64];
    int blk = blockIdx.x;
    int b = blk / N_, n = blk % N_;
    int t = n / P_, p = n % P_;
    int tid = threadIdx.x;
    const float* rf = reps + ((size_t)((b * T_ + t) * P_ + p)) * H_;
    const float* rt = reps + ((size_t)((b * T_ + t + 1) * P_ + p)) * H_;
    float vf = rf[tid], vt = rt[tid];
    red[tid] = vf * vf;
    __syncthreads();
    for (int s = 32; s > 0; s >>= 1) { if (tid < s) red[tid] += red[tid + s]; __syncthreads(); }
    float nf = sqrtf(red[0]) + 1e-4f;
    __syncthreads();
    red[tid] = vt * vt;
    __syncthreads();
    for (int s = 32; s > 0; s >>= 1) { if (tid < s) red[tid] += red[tid + s]; __syncthreads(); }
    float nt = sqrtf(red[0]) + 1e-4f;
    size_t off = ((size_t)b * NP_ + n) * 64 + tid;
    xfn[off] = vf / nf;
    xtn[off] = vt / nt;
}

// ---------------- kernel 3: S = thresh(max(Xfn@Xfn^T, Xtn@Xtn^T)), unguarded WMMA ----------------
__global__ __launch_bounds__(256) void k_sim(const float* __restrict__ xfn,
                                             const float* __restrict__ xtn,
                                             float* __restrict__ S) {
    int wave = blockIdx.x * 8 + (threadIdx.x >> 5);   // total B_*NT_*NT_ waves
    int lane = threadIdx.x & 31;
    int b   = wave / (NT_ * NT_);
    int rem = wave % (NT_ * NT_);
    int it = rem / NT_, jt = rem % NT_;
    int m     = lane & 15;
    int khalf = lane >> 4;
    const float* xfA = xfn + ((size_t)b * NP_ + it * 16 + m) * 64;
    const float* xfB = xfn + ((size_t)b * NP_ + jt * 16 + m) * 64;
    const float* xtA = xtn + ((size_t)b * NP_ + it * 16 + m) * 64;
    const float* xtB = xtn + ((size_t)b * NP_ + jt * 16 + m) * 64;
    v8f cf = {}; v8f ct = {};
#pragma unroll
    for (int k = 0; k < 64; k += 4) {
        int kk = k + 2 * khalf;
        v2f af = *(const v2f*)(xfA + kk);
        v2f bf = *(const v2f*)(xfB + kk);
        v2f at = *(const v2f*)(xtA + kk);
        v2f bt = *(const v2f*)(xtB + kk);
        cf = wmma_f32(af, bf, cf);
        ct = wmma_f32(at, bt, ct);
    }
    int col = jt * 16 + (lane & 15);
    float* Sb = S + (size_t)b * NP_ * NP_;
#pragma unroll
    for (int v = 0; v < 8; ++v) {
        int row = it * 16 + v + 8 * khalf;
        float s = fmaxf(cf[v], ct[v]);
        s = (s > THRESH_) ? s : 0.f;
        Sb[(size_t)row * NP_ + col] = s;              // padded rows/cols are naturally 0
    }
}

// ---------------- kernel 4: Y = X @ W^T, full 16x64 strip per wave ----------------
__global__ __launch_bounds__(256) void k_xw(const float* __restrict__ X,
                                            const float* __restrict__ W,
                                            float* __restrict__ Y) {
    int wave = blockIdx.x * 8 + (threadIdx.x >> 5);   // B_*NP_/16 waves
    int lane = threadIdx.x & 31;
    int m     = lane & 15;
    int khalf = lane >> 4;
    const float* Xa = X + (size_t)(wave * 16 + m) * 64;
    const float* W0 = W + (size_t)(0 * 16 + m) * 64;  // B[k][o] = W[o][k]
    const float* W1 = W + (size_t)(1 * 16 + m) * 64;
    const float* W2 = W + (size_t)(2 * 16 + m) * 64;
    const float* W3 = W + (size_t)(3 * 16 + m) * 64;
    v8f c0 = {}, c1 = {}, c2 = {}, c3 = {};
#pragma unroll
    for (int k = 0; k < 64; k += 4) {
        int kk = k + 2 * khalf;
        v2f a  = *(const v2f*)(Xa + kk);
        v2f b0 = *(const v2f*)(W0 + kk);
        v2f b1 = *(const v2f*)(W1 + kk);
        v2f b2 = *(const v2f*)(W2 + kk);
        v2f b3 = *(const v2f*)(W3 + kk);
        c0 = wmma_f32(a, b0, c0);
        c1 = wmma_f32(a, b1, c1);
        c2 = wmma_f32(a, b2, c2);
        c3 = wmma_f32(a, b3, c3);
    }
    int col = lane & 15;
#pragma unroll
    for (int v = 0; v < 8; ++v) {
        size_t row = (size_t)(wave * 16 + v + 8 * khalf) * 64;
        Y[row + col]      = c0[v];
        Y[row + 16 + col] = c1[v];
        Y[row + 32 + col] = c2[v];
        Y[row + 48 + col] = c3[v];
    }
}

// ---------------- kernel 5: per-node attention scores s_src, s_dst ----------------
__global__ __launch_bounds__(64) void k_scores(const float* __restrict__ xw,
                                               const float* __restrict__ asrc,
                                               const float* __restrict__ adst,
                                               float* __restrict__ ssrc,
                                               float* __restrict__ sdst) {
    __shared__ float r1[64], r2[64];
    int blk = blockIdx.x;             // b*N_ + n (real rows)
    int b = blk / N_, n = blk % N_;
    int tid = threadIdx.x;
    float x = xw[((size_t)b * NP_ + n) * 64 + tid];
    r1[tid] = x * asrc[tid];
    r2[tid] = x * adst[tid];
    __syncthreads();
    for (int s = 32; s > 0; s >>= 1) {
        if (tid < s) { r1[tid] += r1[tid + s]; r2[tid] += r2[tid + s]; }
        __syncthreads();
    }
    if (tid == 0) { ssrc[b * NP_ + n] = r1[0]; sdst[b * NP_ + n] = r2[0]; }
}

// ---------------- kernel 6: softmax row stats (max + 1/sum), real j only ----------------
__global__ __launch_bounds__(128) void k_rowstats(const float* __restrict__ ssrc,
                                                  const float* __restrict__ sdst,
                                                  float* __restrict__ mrow,
                                                  float* __restrict__ rden) {
    __shared__ float red[128];
    int blk = blockIdx.x;             // b*N_ + i
    int b = blk / N_, i = blk % N_;
    int tid = threadIdx.x;
    float di = sdst[b * NP_ + i];
    const float* sb = ssrc + (size_t)b * NP_;
    float mx = -3.4e38f;
    for (int j = tid; j < N_; j += 128) mx = fmaxf(mx, lrelu(di + sb[j]));
    red[tid] = mx;
    __syncthreads();
    for (int s = 64; s > 0; s >>= 1) { if (tid < s) red[tid] = fmaxf(red[tid], red[tid + s]); __syncthreads(); }
    float mv = red[0];
    __syncthreads();
    float sum = 0.f;
    for (int j = tid; j < N_; j += 128) sum += __expf(lrelu(di + sb[j]) - mv);
    red[tid] = sum;
    __syncthreads();
    for (int s = 64; s > 0; s >>= 1) { if (tid < s) red[tid] += red[tid + s]; __syncthreads(); }
    if (tid == 0) { mrow[b * NP_ + i] = mv; rden[b * NP_ + i] = 1.f / red[0]; }
}

// ---------------- kernel 7: out = softmax @ xw + bg; 16x64 strip per wave, alpha in-register ----------------
// Pad rows have rden==0 -> zero weights; xw pad rows are zero -> pad-j contributions vanish.
__global__ __launch_bounds__(256) void k_attn(const float* __restrict__ xw,
                                              const float* __restrict__ ssrc,
                                              const float* __restrict__ sdst,
                                              const float* __restrict__ mrow,
                                              const float* __restrict__ rden,
                                              const float* __restrict__ bg,
                                              float* __restrict__ out) {
    int wave = blockIdx.x * 8 + (threadIdx.x >> 5);   // B_*NT_ waves
    int lane = threadIdx.x & 31;
    int b  = wave / NT_;
    int it = wave % NT_;
    int m     = lane & 15;
    int khalf = lane >> 4;
    int ib = b * NP_ + it * 16 + m;
    float di = sdst[ib];
    float mi = mrow[ib];
    float ri = rden[ib];
    const float* sb  = ssrc + (size_t)b * NP_;
    const float* xwb = xw + (size_t)b * NP_ * 64;
    int col = lane & 15;
    v8f c0 = {}, c1 = {}, c2 = {}, c3 = {};
#pragma unroll 4
    for (int j = 0; j < NP_; j += 4) {
        int j0 = j + 2 * khalf;
        int j1 = j0 + 1;
        v2f a;
        a.x = __expf(lrelu(di + sb[j0]) - mi) * ri;
        a.y = __expf(lrelu(di + sb[j1]) - mi) * ri;
        const float* r0 = xwb + (size_t)j0 * 64 + col;
        const float* r1 = xwb + (size_t)j1 * 64 + col;
        v2f b0; b0.x = r0[0];  b0.y = r1[0];
        v2f b1; b1.x = r0[16]; b1.y = r1[16];
        v2f b2; b2.x = r0[32]; b2.y = r1[32];
        v2f b3; b3.x = r0[48]; b3.y = r1[48];
        c0 = wmma_f32(a, b0, c0);
        c1 = wmma_f32(a, b1, c1);
        c2 = wmma_f32(a, b2, c2);
        c3 = wmma_f32(a, b3, c3);
    }
    float g0 = bg[col], g1 = bg[16 + col], g2 = bg[32 + col], g3 = bg[48 + col];
#pragma unroll
    for (int v = 0; v < 8; ++v) {
        size_t row = ((size_t)b * NP_ + it * 16 + v + 8 * khalf) * 64;
        out[row + col]      = c0[v] + g0;
        out[row + 16 + col] = c1[v] + g1;
        out[row + 32 + col] = c2[v] + g2;
        out[row + 48 + col] = c3[v] + g3;
    }
}

// ---------------- kernel 8: head-mix + l2norm + relu ----------------
__global__ __launch_bounds__(64) void k_headmix(const float* __restrict__ x,
                                                const float* __restrict__ hw,
                                                const float* __restrict__ hb,
                                                float* __restrict__ y) {
    __shared__ float red[64];
    __shared__ float xs[64];
    int blk = blockIdx.x;             // b*N_ + n
    int b = blk / N_, n = blk % N_;
    int tid = threadIdx.x;
    size_t off = ((size_t)b * NP_ + n) * 64;
    xs[tid] = x[off + tid];
    __syncthreads();
    int k = tid >> 4, e = tid & 15;
    float acc = hb[k * HD_ + e];
    const float* w = hw + (size_t)(k * HD_ + e) * HD_;
#pragma unroll
    for (int d = 0; d < HD_; ++d) acc += xs[k * HD_ + d] * w[d];
    red[tid] = acc * acc;
    __syncthreads();
    for (int s = 32; s > 0; s >>= 1) { if (tid < s) red[tid] += red[tid + s]; __syncthreads(); }
    float nrm = fmaxf(sqrtf(red[0]), 1e-12f);
    y[off + tid] = fmaxf(acc / nrm, 0.f);
}

// ---------------- kernel 9: agg = S @ node; 16x64 strip per wave ----------------
__global__ __launch_bounds__(256) void k_agg(const float* __restrict__ S,
                                             const float* __restrict__ node,
                                             float* __restrict__ agg) {
    int wave = blockIdx.x * 8 + (threadIdx.x >> 5);   // B_*NT_ waves
    int lane = threadIdx.x & 31;
    int b  = wave / NT_;
    int it = wave % NT_;
    int m     = lane & 15;
    int khalf = lane >> 4;
    const float* Sr = S + (size_t)b * NP_ * NP_ + (size_t)(it * 16 + m) * NP_;
    const float* nb = node + (size_t)b * NP_ * 64;
    int col = lane & 15;
    v8f c0 = {}, c1 = {}, c2 = {}, c3 = {};
#pragma unroll 4
    for (int j = 0; j < NP_; j += 4) {
        int j0 = j + 2 * khalf;
        v2f a = *(const v2f*)(Sr + j0);
        const float* r0 = nb + (size_t)j0 * 64 + col;
        const float* r1 = r0 + 64;
        v2f b0; b0.x = r0[0];  b0.y = r1[0];
        v2f b1; b1.x = r0[16]; b1.y = r1[16];
        v2f b2; b2.x = r0[32]; b2.y = r1[32];
        v2f b3; b3.x = r0[48]; b3.y = r1[48];
        c0 = wmma_f32(a, b0, c0);
        c1 = wmma_f32(a, b1, c1);
        c2 = wmma_f32(a, b2, c2);
        c3 = wmma_f32(a, b3, c3);
    }
#pragma unroll
    for (int v = 0; v < 8; ++v) {
        size_t row = ((size_t)b * NP_ + it * 16 + v + 8 * khalf) * 64;
        agg[row + col]      = c0[v];
        agg[row + 16 + col] = c1[v];
        agg[row + 32 + col] = c2[v];
        agg[row + 48 + col] = c3[v];
    }
}

// ---------------- kernel 10: spectral MLP per node ----------------
__global__ __launch_bounds__(64) void k_spmlp(const float* __restrict__ agg,
                                              const float* __restrict__ Ws1,
                                              const float* __restrict__ bs1,
                                              const float* __restrict__ Ws2,
                                              const float* __restrict__ bs2,
                                              float* __restrict__ sp) {
    __shared__ float xs[64], h1[64];
    int blk = blockIdx.x;             // b*N_ + n
    int b = blk / N_, n = blk % N_;
    int tid = threadIdx.x;
    size_t off = ((size_t)b * NP_ + n) * 64;
    xs[tid] = agg[off + tid];
    __syncthreads();
    float acc = bs1[tid];
    const float* w = Ws1 + tid * 64;
#pragma unroll 8
    for (int k = 0; k < 64; ++k) acc += xs[k] * w[k];
    h1[tid] = fmaxf(acc, 0.f);
    __syncthreads();
    float acc2 = bs2[tid];
    const float* w2 = Ws2 + tid * 64;
#pragma unroll 8
    for (int k = 0; k < 64; ++k) acc2 += h1[k] * w2[k];
    sp[off + tid] = acc2;
}

// ---------------- kernel 11: means + final classifier ----------------
__global__ __launch_bounds__(192) void k_final(const float* __restrict__ out1,
                                               const float* __restrict__ out2,
                                               const float* __restrict__ sp,
                                               const float* __restrict__ Wf,
                                               const float* __restrict__ bf,
                                               float* __restrict__ out) {
    __shared__ float emb[192];
    int b   = blockIdx.x;
    int tid = threadIdx.x;
    const float* src = (tid < 64) ? out1 : (tid < 128 ? out2 : sp);
    int c = tid & 63;
    float sum = 0.f;
    const float* base = src + ((size_t)b * NP_) * 64 + c;
    for (int n = 0; n < N_; ++n) sum += base[(size_t)n * 64];
    emb[tid] = sum * (1.f / (float)N_);
    __syncthreads();
    if (tid < NC_) {
        float acc = bf[tid];
        const float* w = Wf + tid * 192;
        for (int k = 0; k < 192; ++k) acc += emb[k] * w[k];
        out[b * NC_ + tid] = acc;
    }
}

// ---------------- launcher ----------------
extern "C" void kernel_launch(void* const* d_in, const int* in_sizes, int n_in,
                              void* d_out, int out_size, void* d_ws, size_t ws_size,
                              hipStream_t stream) {
    (void)in_sizes; (void)n_in; (void)out_size; (void)ws_size;
    const float* reps = (const float*)d_in[0];
    const float* Wm1  = (const float*)d_in[1];
    const float* bm1  = (const float*)d_in[2];
    const float* Wm2  = (const float*)d_in[3];
    const float* bm2  = (const float*)d_in[4];
    const float* Wg1  = (const float*)d_in[5];
    const float* a1s  = (const float*)d_in[6];
    const float* a1d  = (const float*)d_in[7];
    const float* bg1  = (const float*)d_in[8];
    const float* hw1  = (const float*)d_in[9];
    const float* hb1  = (const float*)d_in[10];
    const float* Wg2  = (const float*)d_in[11];
    const float* a2s  = (const float*)d_in[12];
    const float* a2d  = (const float*)d_in[13];
    const float* bg2  = (const float*)d_in[14];
    const float* hw3  = (const float*)d_in[15];
    const float* hb3  = (const float*)d_in[16];
    const float* Ws1  = (const float*)d_in[17];
    const float* bs1  = (const float*)d_in[18];
    const float* Ws2  = (const float*)d_in[19];
    const float* bs2  = (const float*)d_in[20];
    const float* Wf   = (const float*)d_in[21];
    const float* bf   = (const float*)d_in[22];
    float* out = (float*)d_out;

    float* w = (float*)d_ws;
    size_t o = 0;
    const size_t RB = (size_t)B_ * NP_ * 64;       // padded row-buffer size
    float* node = w + o; o += RB;
    float* xfn  = w + o; o += RB;
    float* xtn  = w + o; o += RB;
    float* S    = w + o; o += (size_t)B_ * NP_ * NP_;
    float* xw   = w + o; o += RB;
    float* gout = w + o; o += RB;
    float* out1 = w + o; o += RB;
    float* out2 = w + o; o += RB;
    float* agg  = w + o; o += RB;
    float* sp   = w + o; o += RB;
    float* ssrc = w + o; o += (size_t)B_ * NP_;
    float* sdst = w + o; o += (size_t)B_ * NP_;
    float* mrow = w + o; o += (size_t)B_ * NP_;
    float* rden = w + o; o += (size_t)B_ * NP_;

    const int initBlocks = (B_ * (NP_ - N_) * 64 + 255) / 256;  // also covers B_*NP_ stats
    k_initpad<<<initBlocks, 256, 0, stream>>>(node, xfn, xtn, out1, ssrc, sdst, mrow, rden);

    k_node<<<BN_, 64, 0, stream>>>(reps, Wm1, bm1, Wm2, bm2, node);
    k_norm<<<BN_, 64, 0, stream>>>(reps, xfn, xtn);
    k_sim<<<(B_ * NT_ * NT_) / 8, 256, 0, stream>>>(xfn, xtn, S);

    const int xwBlocks   = (B_ * NP_ / 16) / 8;    // 352
    const int attnBlocks = (B_ * NT_) / 8;         // 352

    // GAT layer 1
    k_xw<<<xwBlocks, 256, 0, stream>>>(node, Wg1, xw);
    k_scores<<<BN_, 64, 0, stream>>>(xw, a1s, a1d, ssrc, sdst);
    k_rowstats<<<BN_, 128, 0, stream>>>(ssrc, sdst, mrow, rden);
    k_attn<<<attnBlocks, 256, 0, stream>>>(xw, ssrc, sdst, mrow, rden, bg1, gout);
    k_headmix<<<BN_, 64, 0, stream>>>(gout, hw1, hb1, out1);

    // GAT layer 2
    k_xw<<<xwBlocks, 256, 0, stream>>>(out1, Wg2, xw);
    k_scores<<<BN_, 64, 0, stream>>>(xw, a2s, a2d, ssrc, sdst);
    k_rowstats<<<BN_, 128, 0, stream>>>(ssrc, sdst, mrow, rden);
    k_attn<<<attnBlocks, 256, 0, stream>>>(xw, ssrc, sdst, mrow, rden, bg2, gout);
    k_headmix<<<BN_, 64, 0, stream>>>(gout, hw3, hb3, out2);

    // Spectral branch
    k_agg<<<attnBlocks, 256, 0, stream>>>(S, node, agg);
    k_spmlp<<<BN_, 64, 0, stream>>>(agg, Ws1, bs1, Ws2, bs2, sp);

    k_final<<<B_, 192, 0, stream>>>(out1, out2, sp, Wf, bf, out);
}